// sLSTM_343597384302
// MI455X (gfx1250) — compile-verified
//
#include <hip/hip_runtime.h>

// sLSTM (B=256,T=512,D=8,H=120,L=4, 8 heads x 15) for MI455X / gfx1250.
// Latency-bound sequential scan: 16 workgroups (one per 16-row batch tile),
// 8 waves each (one per head). All matmuls on v_wmma_f32_16x16x4_f32 chains;
// state lives in WMMA accumulator layout; h transposed D->A layout via LDS;
// 120->8 down-projection fused via per-head WMMA partials + ds_add_f32.

#define B_   256
#define T_   512
#define D_   8
#define H_   120
#define L_   4
#define NH_  8
#define HD_  15
#define EPS_ 1e-5f

typedef __attribute__((ext_vector_type(2))) float v2f;
typedef __attribute__((ext_vector_type(8))) float v8f;

__device__ __forceinline__ v8f wmma4(v2f a, v2f b, v8f c) {
  // D = A(16x4,f32) * B(4x16,f32) + C(16x16,f32)
  return __builtin_amdgcn_wmma_f32_16x16x4_f32(
      /*neg_a=*/false, a, /*neg_b=*/false, b,
      /*c_mod=*/(short)0, c, /*reuse_a=*/false, /*reuse_b=*/false);
}

__device__ __forceinline__ float fast_sigmoid(float x) {
  return 1.f / (1.f + __expf(-x));
}
__device__ __forceinline__ float fast_tanh(float x) {
  x = fminf(fmaxf(x, -15.f), 15.f);     // avoid inf/inf
  float e = __expf(2.f * x);
  return (e - 1.f) / (e + 1.f);
}

__global__ __launch_bounds__(256) void slstm_scan(
    const float* __restrict__ x_in, float* __restrict__ x_out,
    const float* __restrict__ Wi, const float* __restrict__ Wf,
    const float* __restrict__ Wz, const float* __restrict__ Wo,
    const float* __restrict__ Ri, const float* __restrict__ Rf,
    const float* __restrict__ Rz, const float* __restrict__ Ro,
    const float* __restrict__ bi, const float* __restrict__ bf,
    const float* __restrict__ bz, const float* __restrict__ bo,
    const float* __restrict__ Wp, const float* __restrict__ bp,
    int layer)
{
  // per-wave 16x16 h tile, row stride 18 floats (bank pad + keeps b64 loads 8B aligned)
  __shared__ float lds_h[8 * 16 * 18];
  __shared__ float y_lds[16 * D_];     // per-step projection accumulator [M=16][d=8]

  const int tid  = threadIdx.x;
  const int lane = tid & 31;
  const int wid  = tid >> 5;           // head index 0..7
  const int half = lane >> 4;          // lane 16-31 half
  const int nl   = lane & 15;
  const int bt   = blockIdx.x;         // batch tile: rows bt*16 .. bt*16+15

  const float* Wg[4] = {Wi, Wf, Wz, Wo};
  const float* Rg[4] = {Ri, Rf, Rz, Ro};
  const float* bg[4] = {bi, bf, bz, bo};

  // ---- Preload all B-matrix operands (held in VGPRs for the whole scan) ----
  // B layout (K x N, 4x16): VGPR r, lanes0-15: K=r,N=lane ; lanes16-31: K=r+2,N=lane-16
  v2f Bx[4][2];   // input projection: B[k][N] = W[unit N][k], K total = 8
  v2f Br[4][4];   // recurrence:       B[k][N] = R[head][k][N], K padded 15->16
  v2f Bp[4];      // down-projection:  B[k][N] = Wp[d=N][head*15+k], N padded 8->16
  float bias[4];
  #pragma unroll
  for (int g = 0; g < 4; ++g) {
    #pragma unroll
    for (int kc = 0; kc < 2; ++kc)
      #pragma unroll
      for (int r = 0; r < 2; ++r) {
        int K = kc * 4 + r + 2 * half;
        Bx[g][kc][r] = (nl < HD_)
            ? Wg[g][((size_t)layer * H_ + wid * HD_ + nl) * D_ + K] : 0.f;
      }
    #pragma unroll
    for (int kc = 0; kc < 4; ++kc)
      #pragma unroll
      for (int r = 0; r < 2; ++r) {
        int K = kc * 4 + r + 2 * half;
        Br[g][kc][r] = (nl < HD_ && K < HD_)
            ? Rg[g][(((size_t)layer * NH_ + wid) * HD_ + K) * HD_ + nl] : 0.f;
      }
    bias[g] = (nl < HD_) ? bg[g][layer * H_ + wid * HD_ + nl] : 0.f;
  }
  #pragma unroll
  for (int kc = 0; kc < 4; ++kc)
    #pragma unroll
    for (int r = 0; r < 2; ++r) {
      int K = kc * 4 + r + 2 * half;
      Bp[kc][r] = (nl < D_ && K < HD_)
          ? Wp[((size_t)layer * D_ + nl) * H_ + wid * HD_ + K] : 0.f;
    }

  // ---- State in WMMA C/D layout: VGPR j -> (batch row M=j+8*half, unit N=nl) ----
  v8f c_st = {0,0,0,0,0,0,0,0};
  v8f n_st = {0,0,0,0,0,0,0,0};
  v8f m_st = {0,0,0,0,0,0,0,0};
  // h in A layout: A[kc] VGPR r, element (M=nl, K=kc*4+r+2*half)
  v2f Ah[4] = {{0.f,0.f},{0.f,0.f},{0.f,0.f},{0.f,0.f}};

  if (tid < 16 * D_) y_lds[tid] = 0.f;
  __syncthreads();

  const int hbase = wid * (16 * 18);
  const float* xrow = x_in + (size_t)(bt * 16 + nl) * T_ * D_;
  float* orow_base  = x_out + (size_t)(bt * 16 + nl) * T_ * D_;

  for (int t = 0; t < T_; ++t) {
    // x_t in A layout: K = {0,1,4,5}+2*half across the two v2f regs
    v2f Ax0 = *(const v2f*)(xrow + t * D_ + 2 * half);
    v2f Ax1 = *(const v2f*)(xrow + t * D_ + 4 + 2 * half);
    if (t + 1 < T_) __builtin_prefetch(xrow + (t + 1) * D_, 0, 1);

    // ---- Gate pre-activations: bias + x@W^T + h@R, 6 chained WMMAs per gate ----
    v8f acc[4];
    #pragma unroll
    for (int g = 0; g < 4; ++g) {
      v8f a;
      #pragma unroll
      for (int j = 0; j < 8; ++j) a[j] = bias[g];
      a = wmma4(Ax0, Bx[g][0], a);
      a = wmma4(Ax1, Bx[g][1], a);
      #pragma unroll
      for (int kc = 0; kc < 4; ++kc) a = wmma4(Ah[kc], Br[g][kc], a);
      acc[g] = a;
    }

    // ---- Exp-gated state update (in accumulator layout) ----
    v8f h_new;
    #pragma unroll
    for (int j = 0; j < 8; ++j) {
      float it = acc[0][j], ft = acc[1][j];
      float z  = fast_tanh(acc[2][j]);
      float o  = fast_sigmoid(acc[3][j]);
      float mo = m_st[j];
      float mn = fmaxf(ft + mo, it);
      float ip = __expf(it - mn);
      float fp = __expf(ft + mo - mn);
      float cn = fp * c_st[j] + ip * z;
      float nn = fp * n_st[j] + ip;
      c_st[j] = cn; n_st[j] = nn; m_st[j] = mn;
      h_new[j] = o * (cn / nn);          // padded unit 15 stays exactly 0
    }

    // ---- Transpose h: D layout -> A layout via own LDS tile (in-order DS) ----
    #pragma unroll
    for (int j = 0; j < 8; ++j)
      lds_h[hbase + (j + 8 * half) * 18 + nl] = h_new[j];
    #pragma unroll
    for (int kc = 0; kc < 4; ++kc) {
      int K0 = kc * 4 + 2 * half;                       // even -> 8B aligned
      Ah[kc] = *(const v2f*)&lds_h[hbase + nl * 18 + K0];
    }

    // ---- Fused down-projection partial: y += h_head @ Wp_head^T ----
    v8f y = {0,0,0,0,0,0,0,0};
    #pragma unroll
    for (int kc = 0; kc < 4; ++kc) y = wmma4(Ah[kc], Bp[kc], y);
    #pragma unroll
    for (int j = 0; j < 8; ++j)
      if (nl < D_) atomicAdd(&y_lds[(j + 8 * half) * D_ + nl], y[j]);

    __syncthreads();                     // all heads' partials visible
    if (wid == 0) {
      // wave 0's Ax regs hold all 16x8 x_t values exactly once
      float vals[4] = {Ax0.x, Ax0.y, Ax1.x, Ax1.y};
      int   cols[4] = {2*half, 2*half + 1, 4 + 2*half, 5 + 2*half};
      float* orow = orow_base + (size_t)t * D_;
      #pragma unroll
      for (int q = 0; q < 4; ++q) {
        int Nc = cols[q];
        int idx = nl * D_ + Nc;
        orow[Nc] = y_lds[idx] + vals[q] + bp[layer * D_ + Nc];  // residual
        y_lds[idx] = 0.f;                // re-zero for next step
      }
    }
    __syncthreads();                     // zeroing visible before next adds
  }
}

// Final head: batchnorm over batch of last timestep + sigmoid(linear)
__global__ __launch_bounds__(256) void bn_head(
    const float* __restrict__ xf, const float* __restrict__ gamma,
    const float* __restrict__ beta, const float* __restrict__ Wfc,
    const float* __restrict__ bfc, float* __restrict__ out)
{
  __shared__ float s[D_ * B_];
  __shared__ float mu_s[D_], rs_s[D_];
  int b = threadIdx.x;
  float v[D_];
  const float* row = xf + ((size_t)b * T_ + (T_ - 1)) * D_;
  #pragma unroll
  for (int d = 0; d < D_; ++d) { v[d] = row[d]; s[d * B_ + b] = v[d]; }
  __syncthreads();
  if (b < D_) {
    float m = 0.f;
    for (int i = 0; i < B_; ++i) m += s[b * B_ + i];
    m /= (float)B_;
    float var = 0.f;
    for (int i = 0; i < B_; ++i) { float dd = s[b * B_ + i] - m; var += dd * dd; }
    var /= (float)B_;
    mu_s[b] = m; rs_s[b] = rsqrtf(var + EPS_);
  }
  __syncthreads();
  float acc = bfc[0];
  #pragma unroll
  for (int d = 0; d < D_; ++d) {
    float bn = (v[d] - mu_s[d]) * rs_s[d] * gamma[d] + beta[d];
    acc += bn * Wfc[d];
  }
  out[b] = 1.f / (1.f + __expf(-acc));
}

extern "C" void kernel_launch(void* const* d_in, const int* in_sizes, int n_in,
                              void* d_out, int out_size, void* d_ws, size_t ws_size,
                              hipStream_t stream) {
  const float* x     = (const float*)d_in[0];
  const float* Wi    = (const float*)d_in[1];
  const float* Wf    = (const float*)d_in[2];
  const float* Wz    = (const float*)d_in[3];
  const float* Wo    = (const float*)d_in[4];
  const float* Ri    = (const float*)d_in[5];
  const float* Rf    = (const float*)d_in[6];
  const float* Rz    = (const float*)d_in[7];
  const float* Ro    = (const float*)d_in[8];
  const float* bi    = (const float*)d_in[9];
  const float* bf    = (const float*)d_in[10];
  const float* bz    = (const float*)d_in[11];
  const float* bo    = (const float*)d_in[12];
  const float* Wp    = (const float*)d_in[13];
  const float* bp    = (const float*)d_in[14];
  const float* gamma = (const float*)d_in[15];
  const float* beta  = (const float*)d_in[16];
  const float* Wfc   = (const float*)d_in[17];
  const float* bfc   = (const float*)d_in[18];

  float* xA = (float*)d_ws;                       // 4 MB
  float* xB = xA + (size_t)B_ * T_ * D_;          // 4 MB (ws needs >= 8 MB)

  const float* cur = x;
  float* nxt = xA;
  for (int l = 0; l < L_; ++l) {
    slstm_scan<<<B_ / 16, 256, 0, stream>>>(cur, nxt,
        Wi, Wf, Wz, Wo, Ri, Rf, Rz, Ro, bi, bf, bz, bo, Wp, bp, l);
    cur = nxt;
    nxt = (nxt == xA) ? xB : xA;
  }
  bn_head<<<1, 256, 0, stream>>>(cur, gamma, beta, Wfc, bfc, (float*)d_out);
}